// PhysicsInformedLoss_3796751090304
// MI455X (gfx1250) — compile-verified
//
#include <hip/hip_runtime.h>

typedef __attribute__((ext_vector_type(2))) float v2f;
typedef __attribute__((ext_vector_type(8))) float v8f;

#define H 1024
#define W 1024
#define NB 16
#define ROWS_PER_BLOCK 16
#define THREADS 256                      // 256 threads * float4 = 1024 = W
#define NBLK (NB * (H / ROWS_PER_BLOCK)) // 1024 partial sums

__device__ __forceinline__ float4 ld4(const float* __restrict__ p, int y, int x) {
    return *(const float4*)(p + (size_t)y * W + x);
}

// ---------------------------------------------------------------------------
// Stage 1: streaming 5-point reflect stencil + |residual|^2 partial sums.
// One block = one batch x 16-row strip. Rolling up/cur/dn registers so every
// row is fetched from HBM exactly once (compulsory-traffic roofline ~5.5us).
// ---------------------------------------------------------------------------
__global__ __launch_bounds__(THREADS) void pil_stage1(
    const float* __restrict__ pred,    // (16, 2, 1024, 1024) f32
    const float* __restrict__ params,  // (16, 10) f32
    float* __restrict__ partial)       // (NBLK,) f32
{
    const int t     = threadIdx.x;
    const int blk   = blockIdx.x;
    const int b     = blk / (H / ROWS_PER_BLOCK);
    const int chunk = blk % (H / ROWS_PER_BLOCK);
    const int y0    = chunk * ROWS_PER_BLOCK;
    const int x0    = t * 4;

    // coef = 2*K0^2/N0 * 1e-8/(vis + 1e-6), matching the reference's f32 math.
    constexpr double K0d = 2.0 * 3.14159265358979323846 / 1.55e-6;
    const float vis  = params[b * 10 + 1];
    const float coef = (float)(2.0 * K0d * K0d / 1.0) * (1e-8f / (vis + 1e-6f));
    const float cm4  = coef - 4.0f;  // residual = up+dn+l+r + (coef-4)*center

    const float* __restrict__ fr = pred + ((size_t)b * 2 + 0) * (size_t)H * W;
    const float* __restrict__ fi = pred + ((size_t)b * 2 + 1) * (size_t)H * W;

    // reflect-pad column neighbors (pad = index 1 / index W-2)
    const int xl = (x0 == 0)     ? 1     : x0 - 1;
    const int xr = (x0 + 4 == W) ? W - 2 : x0 + 4;

    int y = y0;
    const int yu0 = (y0 == 0) ? 1 : y0 - 1;  // reflect-pad row above
    float4 upR = ld4(fr, yu0, x0), upI = ld4(fi, yu0, x0);
    float4 cuR = ld4(fr, y0,  x0), cuI = ld4(fi, y0,  x0);

    float acc = 0.0f;
    #pragma unroll 4
    for (int r = 0; r < ROWS_PER_BLOCK; ++r, ++y) {
        const int ydn = (y == H - 1) ? H - 2 : y + 1;     // reflect-pad below
        const float4 dnR = ld4(fr, ydn, x0);
        const float4 dnI = ld4(fi, ydn, x0);

        // prefetch two rows ahead (global_prefetch_b8, streaming hint)
        const int ypf = (y + 2 < H) ? y + 2 : H - 1;
        __builtin_prefetch(fr + (size_t)ypf * W + x0, 0, 1);
        __builtin_prefetch(fi + (size_t)ypf * W + x0, 0, 1);

        const size_t row = (size_t)y * W;
        const float lR = fr[row + xl], rR = fr[row + xr];
        const float lI = fi[row + xl], rI = fi[row + xr];

        // real channel
        float s;
        s = upR.x + dnR.x + lR    + cuR.y + cm4 * cuR.x; acc = fmaf(s, s, acc);
        s = upR.y + dnR.y + cuR.x + cuR.z + cm4 * cuR.y; acc = fmaf(s, s, acc);
        s = upR.z + dnR.z + cuR.y + cuR.w + cm4 * cuR.z; acc = fmaf(s, s, acc);
        s = upR.w + dnR.w + cuR.z + rR    + cm4 * cuR.w; acc = fmaf(s, s, acc);
        // imag channel
        s = upI.x + dnI.x + lI    + cuI.y + cm4 * cuI.x; acc = fmaf(s, s, acc);
        s = upI.y + dnI.y + cuI.x + cuI.z + cm4 * cuI.y; acc = fmaf(s, s, acc);
        s = upI.z + dnI.z + cuI.y + cuI.w + cm4 * cuI.z; acc = fmaf(s, s, acc);
        s = upI.w + dnI.w + cuI.z + rI    + cm4 * cuI.w; acc = fmaf(s, s, acc);

        upR = cuR; upI = cuI;
        cuR = dnR; cuI = dnI;
    }

    // wave32 intra-wave reduction
    #pragma unroll
    for (int off = 16; off > 0; off >>= 1)
        acc += __shfl_down(acc, off, 32);

    __shared__ float sm[THREADS / 32];
    if ((t & 31) == 0) sm[t >> 5] = acc;
    __syncthreads();
    if (t == 0) {
        float s = 0.0f;
        #pragma unroll
        for (int i = 0; i < THREADS / 32; ++i) s += sm[i];
        partial[blk] = s;
    }
}

// ---------------------------------------------------------------------------
// Stage 2: single wave32 reduces the 1024 partials with V_WMMA_F32_16X16X4_F32.
// A = ones(16x4)  =>  D[m,n] = sum_k B[k,n] + C[m,n]: each issue folds 64 f32
// partials into the f32 accumulator at full precision. Whole wave active =>
// EXEC all-1s as WMMA requires.
// ---------------------------------------------------------------------------
__global__ __launch_bounds__(32) void pil_stage2(
    const float* __restrict__ partial, float* __restrict__ out, float scale)
{
    const int lane = threadIdx.x;

    v2f a; a.x = 1.0f; a.y = 1.0f;   // all-ones A matrix (16x4 f32)
    v8f c = {};                      // f32 accumulator (16x16)

    #pragma unroll
    for (int base = 0; base < NBLK; base += 64) {
        v2f bv;
        bv.x = partial[base + lane];
        bv.y = partial[base + 32 + lane];
        // 8 args: (neg_a, A, neg_b, B, c_mod, C, reuse_a, reuse_b)
        c = __builtin_amdgcn_wmma_f32_16x16x4_f32(
                false, a, false, bv, (short)0, c, false, false);
    }

    // c[0] on lanes 0..15 holds the 16 column sums (row M=0); fold them.
    float v = c[0];
    v += __shfl_xor(v, 1, 32);
    v += __shfl_xor(v, 2, 32);
    v += __shfl_xor(v, 4, 32);
    v += __shfl_xor(v, 8, 32);
    if (lane == 0) out[0] = v * scale;
}

extern "C" void kernel_launch(void* const* d_in, const int* in_sizes, int n_in,
                              void* d_out, int out_size, void* d_ws, size_t ws_size,
                              hipStream_t stream) {
    const float* pred   = (const float*)d_in[0];  // (16,2,1024,1024) f32
    const float* params = (const float*)d_in[1];  // (16,10) f32
    float* out     = (float*)d_out;
    float* partial = (float*)d_ws;                // NBLK * 4 B = 4 KB scratch

    pil_stage1<<<NBLK, THREADS, 0, stream>>>(pred, params, partial);

    const float scale = 1.0f / (float)((size_t)NB * H * W);  // mean over 16*1024*1024
    pil_stage2<<<1, 32, 0, stream>>>(partial, out, scale);
}